// TDLayer_30090540876459
// MI455X (gfx1250) — compile-verified
//
#include <hip/hip_runtime.h>

// Problem constants (match reference)
#define B_SZ   4
#define N_PTS  8192
#define CIN    64
#define COUT   128
#define KNN    16
#define S_OUT  4096          // N/2
#define MTOT   (B_SZ * S_OUT * KNN)
#define BN_EPS 1e-5f

typedef __attribute__((ext_vector_type(2))) float v2f;
typedef __attribute__((ext_vector_type(8))) float v8f;

// ---------------------------------------------------------------------------
// FPS: one block (1024 threads) per batch. dist[] lives in LDS (32KB).
// Serial farthest-point selection with block-wide argmax (first-index ties,
// matching jnp.argmax). Writes new_xyz output [B,3,S] directly.
// ---------------------------------------------------------------------------
__global__ __launch_bounds__(1024) void fps_kernel(const float* __restrict__ xyz,
                                                   float* __restrict__ out_newxyz,
                                                   int* __restrict__ fps_idx) {
  const int b = blockIdx.x;
  const float* px = xyz + (size_t)b * 3 * N_PTS;
  const float* py = px + N_PTS;
  const float* pz = py + N_PTS;
  __shared__ float dist[N_PTS];
  __shared__ float rv[1024];
  __shared__ int   ri[1024];
  const int t = threadIdx.x;
  const float x0 = px[0], y0 = py[0], z0 = pz[0];
  for (int n = t; n < N_PTS; n += 1024) {
    const float dx = px[n] - x0, dy = py[n] - y0, dz = pz[n] - z0;
    dist[n] = dx * dx + dy * dy + dz * dz;
  }
  if (t == 0) {
    fps_idx[b * S_OUT] = 0;
    out_newxyz[(b * 3 + 0) * S_OUT] = x0;
    out_newxyz[(b * 3 + 1) * S_OUT] = y0;
    out_newxyz[(b * 3 + 2) * S_OUT] = z0;
  }
  __syncthreads();
  for (int s = 1; s < S_OUT; ++s) {
    float bv = -1.0f; int bi = 0x7fffffff;
    for (int n = t; n < N_PTS; n += 1024) {
      const float v = dist[n];
      if (v > bv || (v == bv && n < bi)) { bv = v; bi = n; }
    }
    rv[t] = bv; ri[t] = bi;
    __syncthreads();
    for (int off = 512; off > 0; off >>= 1) {
      if (t < off) {
        const float ov = rv[t + off]; const int oi = ri[t + off];
        if (ov > rv[t] || (ov == rv[t] && oi < ri[t])) { rv[t] = ov; ri[t] = oi; }
      }
      __syncthreads();
    }
    const int idx = ri[0];
    const float qx = px[idx], qy = py[idx], qz = pz[idx];
    if (t == 0) {
      fps_idx[b * S_OUT + s] = idx;
      out_newxyz[(b * 3 + 0) * S_OUT + s] = qx;
      out_newxyz[(b * 3 + 1) * S_OUT + s] = qy;
      out_newxyz[(b * 3 + 2) * S_OUT + s] = qz;
    }
    for (int n = t; n < N_PTS; n += 1024) {
      const float dx = px[n] - qx, dy = py[n] - qy, dz = pz[n] - qz;
      dist[n] = fminf(dist[n], dx * dx + dy * dy + dz * dz);
    }
    __syncthreads();
  }
}

// ---------------------------------------------------------------------------
// kNN: one wave32 per query (8 queries per 256-thread block). Each lane keeps
// a sorted top-16 (fully unrolled insertion, stays in registers), then a
// 16-round cross-lane merge via wave argmin (lexicographic (dist, idx) to
// match lax.top_k tie-breaking). Distance uses |q|^2+|p|^2-2 q.p like the ref.
// ---------------------------------------------------------------------------
__global__ __launch_bounds__(256) void knn_kernel(const float* __restrict__ xyz,
                                                  const float* __restrict__ newxyz,
                                                  int* __restrict__ knn_idx) {
  const int lane = threadIdx.x & 31;
  const int qi = blockIdx.x * 8 + (threadIdx.x >> 5);
  const int b = qi / S_OUT, s = qi - b * S_OUT;
  const float* px = xyz + (size_t)b * 3 * N_PTS;
  const float* py = px + N_PTS;
  const float* pz = py + N_PTS;
  const float qx = newxyz[(b * 3 + 0) * S_OUT + s];
  const float qy = newxyz[(b * 3 + 1) * S_OUT + s];
  const float qz = newxyz[(b * 3 + 2) * S_OUT + s];
  const float sqq = qx * qx + qy * qy + qz * qz;
  float bd[KNN]; int bi[KNN];
#pragma unroll
  for (int j = 0; j < KNN; ++j) { bd[j] = 3.0e38f; bi[j] = 0x7fffffff; }
  for (int n = lane; n < N_PTS; n += 32) {
    const float x = px[n], y = py[n], z = pz[n];
    const float d = sqq + (x * x + y * y + z * z) - 2.0f * (qx * x + qy * y + qz * z);
    if (d < bd[KNN - 1] || (d == bd[KNN - 1] && n < bi[KNN - 1])) {
      float cd = d; int ci = n;
#pragma unroll
      for (int j = 0; j < KNN; ++j) {
        const bool sw = (cd < bd[j]) || (cd == bd[j] && ci < bi[j]);
        const float td = sw ? bd[j] : cd; const int ti = sw ? bi[j] : ci;
        bd[j] = sw ? cd : bd[j];          bi[j] = sw ? ci : bi[j];
        cd = td; ci = ti;
      }
    }
  }
  int ptr = 0;
  for (int r = 0; r < KNN; ++r) {
    float hv = 3.0e38f; int hi = 0x7fffffff;
#pragma unroll
    for (int j = 0; j < KNN; ++j) if (ptr == j) { hv = bd[j]; hi = bi[j]; }
    float mv = hv; int mi = hi; int ml = lane;
#pragma unroll
    for (int off = 16; off > 0; off >>= 1) {
      const float ov = __shfl_xor(mv, off, 32);
      const int   oi = __shfl_xor(mi, off, 32);
      const int   ol = __shfl_xor(ml, off, 32);
      if (ov < mv || (ov == mv && oi < mi)) { mv = ov; mi = oi; ml = ol; }
    }
    if (lane == ml) ptr++;
    if (lane == 0) knn_idx[qi * KNN + r] = mi;
  }
}

// ---------------------------------------------------------------------------
// 16x16 output tile via 16 chained V_WMMA_F32_16X16X4_F32 over the CIN=64
// reduction. A rows are the 16 neighborhood points (from LDS), B columns are
// output channels (weights read directly; L2-resident). Bias folded into C.
// VGPR layouts per ISA 7.12.2: lane<16 -> M/N = lane, K-offset 0/1;
// lanes 16-31 -> K-offset 2/3.  C/D: lane<16 rows 0-7, lanes>=16 rows 8-15.
// ---------------------------------------------------------------------------
__device__ __forceinline__ v8f gemm_tile_16x16(const float* As, const float* __restrict__ W,
                                               float bias, int n, int mr, int half) {
  v8f acc;
#pragma unroll
  for (int r = 0; r < 8; ++r) acc[r] = bias;
#pragma unroll
  for (int kk = 0; kk < CIN; kk += 4) {
    v2f a, bb;
    a.x  = As[mr * CIN + kk + 2 * half];
    a.y  = As[mr * CIN + kk + 2 * half + 1];
    bb.x = W[n * CIN + kk + 2 * half];
    bb.y = W[n * CIN + kk + 2 * half + 1];
    acc = __builtin_amdgcn_wmma_f32_16x16x4_f32(false, a, false, bb, (short)0, acc,
                                                false, false);
  }
  return acc;
}

__device__ __forceinline__ void stage_group(const float* __restrict__ points,
                                            const int* __restrict__ knn_idx,
                                            int g, int b, int tid,
                                            int* gi, float* As) {
  if (tid < KNN) gi[tid] = knn_idx[g * KNN + tid];
  __syncthreads();
  const float* pb = points + (size_t)b * CIN * N_PTS;
  for (int e = tid; e < KNN * CIN; e += 128) {
    const int k = e >> 6, c = e & (CIN - 1);
    As[k * CIN + c] = pb[(size_t)c * N_PTS + gi[k]];
  }
  __syncthreads();
}

// Pass 1: conv1 (+bias) and accumulate per-channel sum / sumsq for BN1.
__global__ __launch_bounds__(128) void conv1_stats_kernel(
    const float* __restrict__ points, const int* __restrict__ knn_idx,
    const float* __restrict__ w1, const float* __restrict__ b1,
    float* __restrict__ sum1, float* __restrict__ sq1) {
  const int g = blockIdx.x;
  const int b = g / S_OUT;
  const int tid = threadIdx.x;
  __shared__ int   gi[KNN];
  __shared__ float As[KNN * CIN];
  stage_group(points, knn_idx, g, b, tid, gi, As);
  const int lane = tid & 31, warp = tid >> 5;
  const int half = lane >> 4, mr = lane & 15;
  const int n = warp * 16 + mr;
  const v8f y = gemm_tile_16x16(As, w1, b1[n], n, mr, half);
  float sm = 0.f, sq = 0.f;
#pragma unroll
  for (int r = 0; r < 8; ++r) { sm += y[r]; sq += y[r] * y[r]; }
  sm += __shfl_xor(sm, 16, 32);
  sq += __shfl_xor(sq, 16, 32);
  if (lane < 16) { atomicAdd(&sum1[n], sm); atomicAdd(&sq1[n], sq); }
}

// sum/sumsq -> fused affine BN params: x' = a*x + c
__global__ void bn_params_kernel(const float* __restrict__ sum, const float* __restrict__ sq,
                                 const float* __restrict__ gamma, const float* __restrict__ beta,
                                 float* __restrict__ a, float* __restrict__ c, int C) {
  const int i = threadIdx.x;
  if (i < C) {
    const float inv = 1.0f / (float)MTOT;
    const float m = sum[i] * inv;
    const float v = sq[i] * inv - m * m;
    const float aa = gamma[i] * rsqrtf(v + BN_EPS);
    a[i] = aa;
    c[i] = beta[i] - aa * m;
  }
}

// Pass 2: recompute conv1 -> BN1 -> ReLU (into LDS), conv2 (+bias), BN2 stats.
__global__ __launch_bounds__(128) void conv2_stats_kernel(
    const float* __restrict__ points, const int* __restrict__ knn_idx,
    const float* __restrict__ w1, const float* __restrict__ b1,
    const float* __restrict__ a1, const float* __restrict__ c1,
    const float* __restrict__ w2, const float* __restrict__ b2,
    float* __restrict__ sum2, float* __restrict__ sq2) {
  const int g = blockIdx.x;
  const int b = g / S_OUT;
  const int tid = threadIdx.x;
  __shared__ int   gi[KNN];
  __shared__ float As[KNN * CIN];
  __shared__ float X1[KNN * CIN];
  stage_group(points, knn_idx, g, b, tid, gi, As);
  const int lane = tid & 31, warp = tid >> 5;
  const int half = lane >> 4, mr = lane & 15;
  {
    const int n = warp * 16 + mr;
    const v8f y = gemm_tile_16x16(As, w1, b1[n], n, mr, half);
    const float aa = a1[n], cc = c1[n];
    const int rbase = half * 8;
#pragma unroll
    for (int r = 0; r < 8; ++r)
      X1[(rbase + r) * CIN + n] = fmaxf(0.f, aa * y[r] + cc);
  }
  __syncthreads();
#pragma unroll
  for (int t2 = 0; t2 < 2; ++t2) {
    const int n = (warp + t2 * 4) * 16 + mr;
    const v8f y = gemm_tile_16x16(X1, w2, b2[n], n, mr, half);
    float sm = 0.f, sq = 0.f;
#pragma unroll
    for (int r = 0; r < 8; ++r) { sm += y[r]; sq += y[r] * y[r]; }
    sm += __shfl_xor(sm, 16, 32);
    sq += __shfl_xor(sq, 16, 32);
    if (lane < 16) { atomicAdd(&sum2[n], sm); atomicAdd(&sq2[n], sq); }
  }
}

// Pass 3: recompute conv1->BN1->ReLU->conv2->BN2->ReLU, max-pool over K=16
// (= the 16 tile rows: per-lane max over 8 regs + shfl_xor(16) combine).
__global__ __launch_bounds__(128) void final_kernel(
    const float* __restrict__ points, const int* __restrict__ knn_idx,
    const float* __restrict__ w1, const float* __restrict__ b1,
    const float* __restrict__ a1, const float* __restrict__ c1,
    const float* __restrict__ w2, const float* __restrict__ b2,
    const float* __restrict__ a2, const float* __restrict__ c2,
    float* __restrict__ out_pooled) {
  const int g = blockIdx.x;
  const int b = g / S_OUT;
  const int s = g - b * S_OUT;
  const int tid = threadIdx.x;
  __shared__ int   gi[KNN];
  __shared__ float As[KNN * CIN];
  __shared__ float X1[KNN * CIN];
  stage_group(points, knn_idx, g, b, tid, gi, As);
  const int lane = tid & 31, warp = tid >> 5;
  const int half = lane >> 4, mr = lane & 15;
  {
    const int n = warp * 16 + mr;
    const v8f y = gemm_tile_16x16(As, w1, b1[n], n, mr, half);
    const float aa = a1[n], cc = c1[n];
    const int rbase = half * 8;
#pragma unroll
    for (int r = 0; r < 8; ++r)
      X1[(rbase + r) * CIN + n] = fmaxf(0.f, aa * y[r] + cc);
  }
  __syncthreads();
#pragma unroll
  for (int t2 = 0; t2 < 2; ++t2) {
    const int n = (warp + t2 * 4) * 16 + mr;
    const v8f y = gemm_tile_16x16(X1, w2, b2[n], n, mr, half);
    const float aa = a2[n], cc = c2[n];
    float mx = -3.0e38f;
#pragma unroll
    for (int r = 0; r < 8; ++r) mx = fmaxf(mx, aa * y[r] + cc);
    mx = fmaxf(mx, __shfl_xor(mx, 16, 32));
    if (lane < 16)
      out_pooled[((size_t)b * COUT + n) * S_OUT + s] = fmaxf(mx, 0.f);
  }
}

__global__ void zero_stats_kernel(float* __restrict__ stats) {
  const int t = blockIdx.x * blockDim.x + threadIdx.x;
  if (t < 2 * (CIN + COUT)) stats[t] = 0.f;
}

// ---------------------------------------------------------------------------
// Launch: inputs order = xyz, points, w1, b1, gamma1, beta1, w2, b2, gamma2,
// beta2 (all f32). d_out = [new_xyz B*3*S | pooled B*COUT*S]. Workspace use:
// ~1.2 MB (fps idx, knn idx, stats, fused BN params).
// ---------------------------------------------------------------------------
extern "C" void kernel_launch(void* const* d_in, const int* in_sizes, int n_in,
                              void* d_out, int out_size, void* d_ws, size_t ws_size,
                              hipStream_t stream) {
  const float* xyz    = (const float*)d_in[0];
  const float* points = (const float*)d_in[1];
  const float* w1     = (const float*)d_in[2];
  const float* b1     = (const float*)d_in[3];
  const float* g1     = (const float*)d_in[4];
  const float* be1    = (const float*)d_in[5];
  const float* w2     = (const float*)d_in[6];
  const float* b2     = (const float*)d_in[7];
  const float* g2     = (const float*)d_in[8];
  const float* be2    = (const float*)d_in[9];
  float* out = (float*)d_out;
  float* out_newxyz = out;
  float* out_pooled = out + (size_t)B_SZ * 3 * S_OUT;

  int*   fps_idx = (int*)d_ws;
  int*   knn_idx = fps_idx + B_SZ * S_OUT;
  float* stats   = (float*)(knn_idx + B_SZ * S_OUT * KNN);
  float* sum1 = stats;
  float* sq1  = sum1 + CIN;
  float* sum2 = sq1 + CIN;
  float* sq2  = sum2 + COUT;
  float* a1   = sq2 + COUT;
  float* c1   = a1 + CIN;
  float* a2   = c1 + CIN;
  float* c2   = a2 + COUT;

  fps_kernel<<<B_SZ, 1024, 0, stream>>>(xyz, out_newxyz, fps_idx);
  knn_kernel<<<(B_SZ * S_OUT) / 8, 256, 0, stream>>>(xyz, out_newxyz, knn_idx);
  zero_stats_kernel<<<1, 512, 0, stream>>>(stats);
  conv1_stats_kernel<<<B_SZ * S_OUT, 128, 0, stream>>>(points, knn_idx, w1, b1, sum1, sq1);
  bn_params_kernel<<<1, 128, 0, stream>>>(sum1, sq1, g1, be1, a1, c1, CIN);
  conv2_stats_kernel<<<B_SZ * S_OUT, 128, 0, stream>>>(points, knn_idx, w1, b1, a1, c1,
                                                       w2, b2, sum2, sq2);
  bn_params_kernel<<<1, 128, 0, stream>>>(sum2, sq2, g2, be2, a2, c2, COUT);
  final_kernel<<<B_SZ * S_OUT, 128, 0, stream>>>(points, knn_idx, w1, b1, a1, c1,
                                                 w2, b2, a2, c2, out_pooled);
}